// WatermarkLoss_13915694039614
// MI455X (gfx1250) — compile-verified
//
#include <hip/hip_runtime.h>

typedef __attribute__((ext_vector_type(2))) float v2f;
typedef __attribute__((ext_vector_type(4))) float v4f;
typedef __attribute__((ext_vector_type(8))) float v8f;

#define T_ROWS   128
#define M_COLS   3200000      // 50000 * 64
#define SLICE    1024         // K-elements per workgroup
#define NBLK     (M_COLS / SLICE)   // 3125

// ---------------------------------------------------------------------------
// Stage 0: zero the 128-float partial-logit accumulator in workspace.
// ---------------------------------------------------------------------------
__global__ void wm_zero(float* __restrict__ logits) {
    logits[threadIdx.x] = 0.0f;
}

// ---------------------------------------------------------------------------
// Stage 1: GEMV via V_WMMA_F32_16X16X4_F32.
//   block = 256 threads (8 waves). Wave w owns rows 16w..16w+15.
//   Each block owns a 1024-wide K-slice; 256 WMMAs per wave.
//
//   K-permutation trick: per 8-column group, low lanes (0-15) load float4
//   covering columns k..k+3, hi lanes (16-31) cover k+4..k+7 (one contiguous
//   b128 per lane). WMMA #1 consumes slots (x,y), WMMA #2 slots (z,w); since
//   A and B (conv1 broadcast) use the identical lane<->k mapping and WMMA
//   reduces over K, every column is counted exactly once.
//
//   D column 0 -> lane 0 holds rows m..m+7, lane 16 rows m+8..m+15.
// ---------------------------------------------------------------------------
__global__ __launch_bounds__(256)
void wm_gemv_wmma(const float* __restrict__ X,
                  const float* __restrict__ conv1,
                  float* __restrict__ logits) {
    const int tid  = threadIdx.x;
    const int wave = tid >> 5;          // 0..7  (wave32)
    const int lane = tid & 31;
    const int hi   = (lane >> 4) & 1;   // lanes 16..31 take the upper 4 cols
    const int mrow = lane & 15;         // A-matrix row within tile
    const int row  = wave * 16 + mrow;  // global row of X (0..127)

    const long long base = (long long)blockIdx.x * SLICE;

    const float* __restrict__ pa = X + (long long)row * M_COLS + base + 4 * hi;
    const float* __restrict__ pb = conv1 + base + 4 * hi;

    v8f c = {0.f, 0.f, 0.f, 0.f, 0.f, 0.f, 0.f, 0.f};

#pragma unroll 4
    for (int k = 0; k < SLICE; k += 8) {
        v4f a4 = __builtin_nontemporal_load((const v4f*)(pa + k)); // X: stream once
        v4f b4 = *(const v4f*)(pb + k);                            // conv1: cached
        v2f a0 = {a4.x, a4.y};
        v2f a1 = {a4.z, a4.w};
        v2f b0 = {b4.x, b4.y};
        v2f b1 = {b4.z, b4.w};
        // 8 args: (neg_a, A, neg_b, B, c_mod, C, reuse_a, reuse_b)
        c = __builtin_amdgcn_wmma_f32_16x16x4_f32(
                false, a0, false, b0, (short)0, c, false, false);
        c = __builtin_amdgcn_wmma_f32_16x16x4_f32(
                false, a1, false, b1, (short)0, c, false, false);
    }

    // Column N=0 of D: lane 0 -> rows wave*16+0..7, lane 16 -> rows +8..15.
    if (mrow == 0) {
        const int r0 = wave * 16 + hi * 8;
#pragma unroll
        for (int i = 0; i < 8; ++i)
            atomicAdd(&logits[r0 + i], c[i]);
    }
}

// ---------------------------------------------------------------------------
// Stage 2: sigmoid + BCE, summed over the 128 rows -> scalar loss.
// Mirrors the reference numerics exactly (including fp32 sigmoid saturation
// driving log/log1p to +/-inf), NOT the stable logits form.
// ---------------------------------------------------------------------------
__global__ void wm_bce(const float* __restrict__ logits,
                       const float* __restrict__ b,
                       float* __restrict__ out) {
    __shared__ float s[T_ROWS];
    const int t = threadIdx.x;
    const float z  = logits[t];
    const float y  = 1.0f / (1.0f + expf(-z));
    const float bt = b[t];
    s[t] = -(bt * logf(y) + (1.0f - bt) * log1pf(-y));
    __syncthreads();
    if (t == 0) {
        float acc = 0.0f;
        for (int i = 0; i < T_ROWS; ++i) acc += s[i];
        out[0] = acc;
    }
}

// ---------------------------------------------------------------------------
// Harness entry point. Inputs (setup_inputs order): conv1, b, X.
// ---------------------------------------------------------------------------
extern "C" void kernel_launch(void* const* d_in, const int* in_sizes, int n_in,
                              void* d_out, int out_size, void* d_ws, size_t ws_size,
                              hipStream_t stream) {
    const float* conv1 = (const float*)d_in[0];
    const float* b     = (const float*)d_in[1];
    const float* X     = (const float*)d_in[2];
    float* logits      = (float*)d_ws;          // 128 floats of scratch
    float* out         = (float*)d_out;         // scalar loss

    wm_zero<<<1, T_ROWS, 0, stream>>>(logits);
    wm_gemv_wmma<<<NBLK, 256, 0, stream>>>(X, conv1, logits);
    wm_bce<<<1, T_ROWS, 0, stream>>>(logits, b, out);
}